// FeatureTransformerSlice_17643725651979
// MI455X (gfx1250) — compile-verified
//
#include <hip/hip_runtime.h>

// NNUE feature-transformer slice on MI455X (gfx1250).
//
// out_row(64x16) = S(64x32) . G(32x16) per batch row, computed exactly in f32
// with V_WMMA_F32_16X16X4_F32 (16x4 A, 4x16 B, 16x16 C/D), one wave32 per row.
//   S[sq,m] = val_m * [floor(idx_m/704) == sq]   (one-hot scaled scatter)
//   G[m,:]  = weight[idx_m, :]                    (gather; weight is L2-resident)
//
// 4 M-tiles (16 squares each) x 8 K-steps (4 features each) = 32 WMMA per row.
// Output (67 MB) store bandwidth is the bottleneck (~2.9us @ 23.3 TB/s); the
// WMMA + A-build work is sized to fit underneath it. sq is computed once per
// lane and packed with idx into one word so the per-step broadcasts stay at
// 4 ds_bpermute and unpack is 2 cheap VALU ops instead of re-dividing.

typedef __attribute__((ext_vector_type(2))) float v2f;
typedef __attribute__((ext_vector_type(8))) float v8f;

#define INPUTS_PER_SQUARE 704u
#define OUTPUTS_PER_SQUARE 16
#define NUM_SQUARES 64
#define MAX_ACTIVE 32
#define ROWS_PER_BLOCK 8   // 8 wave32s per 256-thread block

__global__ __launch_bounds__(256) void ftrans_wmma_kernel(
    const int* __restrict__ fidx,
    const float* __restrict__ fval,
    const float* __restrict__ weight,
    float* __restrict__ out,
    int batch)
{
    const int lane = threadIdx.x & 31;
    const int wave = threadIdx.x >> 5;
    const int row  = blockIdx.x * ROWS_PER_BLOCK + wave;
    if (row >= batch) return;   // wave-uniform: EXEC stays all-ones for WMMA

    // Per-lane feature load (lane m <-> feature m of this row).
    int   idx = fidx[(size_t)row * MAX_ACTIVE + lane];
    float val = fval[(size_t)row * MAX_ACTIVE + lane];
    if (idx < 0) { idx = 0; val = 0.0f; }   // -1 marks empty slots

    // One unsigned divide per lane; pack (sq << 16) | idx (idx < 45056 < 2^16).
    const unsigned sq_lane = (unsigned)idx / INPUTS_PER_SQUARE;
    const int packed = (int)((sq_lane << 16) | (unsigned)idx);

    const int half = lane >> 4;     // 0: lanes 0-15, 1: lanes 16-31
    const int n    = lane & 15;     // N column / M-within-halfwave

    v8f acc[4];
    #pragma unroll
    for (int t = 0; t < 4; ++t) acc[t] = (v8f){0,0,0,0,0,0,0,0};

    // K loop: 8 steps of 4 features. A-matrix f32 16x4 layout: lanes 0-15 hold
    // K=0 (vgpr0), K=1 (vgpr1); lanes 16-31 hold K=2,3. B 4x16 mirrors that K
    // split across lane halves, N striped across lanes within a VGPR.
    #pragma unroll
    for (int k = 0; k < 8; ++k) {
        const int f0 = 4 * k + 2 * half;   // this lane's two features this step
        const int f1 = f0 + 1;

        // Wave-level broadcast of (packed, val) for the needed features.
        const int   p0 = __shfl(packed, f0, 32);
        const int   p1 = __shfl(packed, f1, 32);
        const float v0 = __shfl(val, f0, 32);
        const float v1 = __shfl(val, f1, 32);

        const int i0 = p0 & 0xFFFF;                 // weight row index
        const int i1 = p1 & 0xFFFF;
        const int s0 = (int)((unsigned)p0 >> 16);   // square id
        const int s1 = (int)((unsigned)p1 >> 16);

        // B gather: each half-wave reads one contiguous 64B weight row (L2 hit).
        v2f b;
        b.x = weight[(size_t)i0 * OUTPUTS_PER_SQUARE + n];
        b.y = weight[(size_t)i1 * OUTPUTS_PER_SQUARE + n];

        #pragma unroll
        for (int t = 0; t < 4; ++t) {
            const int msq = t * 16 + n;          // A's M row = square within tile
            v2f a;
            a.x = (s0 == msq) ? v0 : 0.0f;
            a.y = (s1 == msq) ? v1 : 0.0f;
            // D = A x B + C, full f32 (exact).
            // (neg_a, A, neg_b, B, c_mod, C, reuse_a, reuse_b)
            acc[t] = __builtin_amdgcn_wmma_f32_16x16x4_f32(
                false, a, false, b, (short)0, acc[t], false, false);
        }
    }

    // Store: C/D f32 16x16 layout -> VGPR v holds M=v (lanes 0-15) / M=v+8
    // (lanes 16-31), N = lane&15. square = 16*t + v + 8*half. Each b32 store
    // writes two contiguous 64B segments (half-wave each): fully coalesced.
    float* __restrict__ orow = out + (size_t)row * (NUM_SQUARES * OUTPUTS_PER_SQUARE);
    #pragma unroll
    for (int t = 0; t < 4; ++t) {
        #pragma unroll
        for (int v = 0; v < 8; ++v) {
            const int sq = t * 16 + v + half * 8;
            orow[sq * OUTPUTS_PER_SQUARE + n] = acc[t][v];
        }
    }
}

extern "C" void kernel_launch(void* const* d_in, const int* in_sizes, int n_in,
                              void* d_out, int out_size, void* d_ws, size_t ws_size,
                              hipStream_t stream) {
    const int*   fidx   = (const int*)d_in[0];     // (B, 32) int32
    const float* fval   = (const float*)d_in[1];   // (B, 32) f32
    const float* weight = (const float*)d_in[2];   // (45056, 16) f32
    float*       out    = (float*)d_out;           // (B, 1024) f32

    const int batch = in_sizes[0] / MAX_ACTIVE;
    const int grid  = (batch + ROWS_PER_BLOCK - 1) / ROWS_PER_BLOCK;
    ftrans_wmma_kernel<<<grid, 256, 0, stream>>>(fidx, fval, weight, out, batch);
}